// MultiBoxLoss_10943576670681
// MI455X (gfx1250) — compile-verified
//
#include <hip/hip_runtime.h>
#include <hip/hip_bf16.h>

// ---------------------------------------------------------------------------
// MultiBoxLoss on MI455X (gfx1250): pure streaming reduction, HBM-bound.
// 44 B/row * 2e6 rows = 88 MB -> ~3.8us floor at 23.3 TB/s.
// Data path: async global->LDS DMA (ASYNCcnt) with double buffering,
// wave32 shuffle reductions, deterministic 2-stage block reduction.
// ---------------------------------------------------------------------------

#define TILE      256          // rows per tile (== blockDim.x)
#define NTHREADS  256          // 8 waves of 32
#define NBLOCKS   1024

// LDS layout per buffer (bytes):
//   [0,2048)      cout  (TILE * 2 f32)
//   [2048,6144)   rout  (TILE * 4 f32)
//   [6144,10240)  reg_t (TILE * 4 f32)
//   [10240,11264) cls_t (TILE * i32)
#define BUF_BYTES 11264

// ---- gfx1250 async global->LDS plumbing --------------------------------
// Probe-confirmed (round 1 stderr): builtins exist with signatures
//   __builtin_amdgcn_global_load_async_to_lds_b32 (AS1 int*       g, AS3 lds, imm off, imm cpol)
//   __builtin_amdgcn_global_load_async_to_lds_b64 (AS1 v2i32*     g, AS3 lds, imm off, imm cpol)
//   __builtin_amdgcn_global_load_async_to_lds_b128(AS1 v4i32*     g, AS3 lds, imm off, imm cpol)
// (global param is non-const, typed to the transfer width, vector_size-style)
typedef int v2i32 __attribute__((vector_size(8)));
typedef int v4i32 __attribute__((vector_size(16)));

#define AS1P(T, p) ((__attribute__((address_space(1))) T*)(void*)(p))
#define AS3P(T, p) ((__attribute__((address_space(3))) T*)(p))

#if defined(__has_builtin)
#if __has_builtin(__builtin_amdgcn_global_load_async_to_lds_b128) && \
    __has_builtin(__builtin_amdgcn_global_load_async_to_lds_b64)  && \
    __has_builtin(__builtin_amdgcn_global_load_async_to_lds_b32)
#define ASYNC_VIA_BUILTIN 1
#endif
#if __has_builtin(__builtin_amdgcn_s_wait_asynccnt)
#define WAIT_ASYNC(n) __builtin_amdgcn_s_wait_asynccnt((unsigned short)(n))
#endif
#endif

#ifndef WAIT_ASYNC
#define WAIT_ASYNC(n) asm volatile("s_wait_asynccnt %0" :: "i"(n) : "memory")
#endif

#if ASYNC_VIA_BUILTIN
#define ASYNC_B32(g, l)                                                      \
  __builtin_amdgcn_global_load_async_to_lds_b32(AS1P(int, (g)),              \
                                                AS3P(int, (l)), 0, 0)
#define ASYNC_B64(g, l)                                                      \
  __builtin_amdgcn_global_load_async_to_lds_b64(AS1P(v2i32, (g)),            \
                                                AS3P(v2i32, (l)), 0, 0)
#define ASYNC_B128(g, l)                                                     \
  __builtin_amdgcn_global_load_async_to_lds_b128(AS1P(v4i32, (g)),           \
                                                 AS3P(v4i32, (l)), 0, 0)
#else
// Low 32 bits of a generic pointer into __shared__ are the LDS byte address
// (flat LDS aperture).
#define LDS_OFF(l) ((unsigned)(unsigned long long)(l))
#define ASYNC_B32(g, l)                                                     \
  asm volatile("global_load_async_to_lds_b32 %0, %1, off"                   \
               :: "v"(LDS_OFF(l)), "v"(g) : "memory")
#define ASYNC_B64(g, l)                                                     \
  asm volatile("global_load_async_to_lds_b64 %0, %1, off"                   \
               :: "v"(LDS_OFF(l)), "v"(g) : "memory")
#define ASYNC_B128(g, l)                                                    \
  asm volatile("global_load_async_to_lds_b128 %0, %1, off"                  \
               :: "v"(LDS_OFF(l)), "v"(g) : "memory")
#endif

// ---- per-row math -------------------------------------------------------
__device__ __forceinline__ void accum_row(float x0, float x1, int lbl,
                                          float4 d, float4 g,
                                          float& cAcc, float& rAcc)
{
  // 2-class log-softmax cross entropy, numerically stable:
  // ce = max + log1p(exp(min-max)) - x[clip(lbl,0,1)]
  float mx  = fmaxf(x0, x1);
  float mn  = fminf(x0, x1);
  float lse = mx + log1pf(__expf(mn - mx));
  float xl  = (lbl == 1) ? x1 : x0;
  if (lbl <= 1) cAcc += lse - xl;          // selected (pos|neg) anchors

  if (lbl == 1) {                          // positive anchors only
    float s = 0.0f, t, a;
    t = d.x - g.x; a = fabsf(t); s += (a < 1.0f) ? 0.5f * t * t : a - 0.5f;
    t = d.y - g.y; a = fabsf(t); s += (a < 1.0f) ? 0.5f * t * t : a - 0.5f;
    t = d.z - g.z; a = fabsf(t); s += (a < 1.0f) ? 0.5f * t * t : a - 0.5f;
    t = d.w - g.w; a = fabsf(t); s += (a < 1.0f) ? 0.5f * t * t : a - 0.5f;
    rAcc += s * 0.25f;
  }
}

// Every thread issues exactly 4 async instructions -> ASYNCcnt advances
// uniformly (+4 per tile per wave), so s_wait_asynccnt thresholds are valid
// for all 8 waves.
__device__ __forceinline__ void issue_tile(const float* __restrict__ cout,
                                           const float* __restrict__ rout,
                                           const int*   __restrict__ cls,
                                           const float* __restrict__ regt,
                                           long long row0,
                                           unsigned char* buf, int tid)
{
  ASYNC_B64 (cout + row0 * 2 + tid * 2, buf +     0 + tid *  8);  // 256*8   = 2048 B
  ASYNC_B128(rout + row0 * 4 + tid * 4, buf +  2048 + tid * 16);  // 256*16  = 4096 B
  ASYNC_B128(regt + row0 * 4 + tid * 4, buf +  6144 + tid * 16);  // 256*16  = 4096 B
  ASYNC_B32 (cls  + row0     + tid,     buf + 10240 + tid *  4);  // 256*4   = 1024 B
}

__global__ __launch_bounds__(NTHREADS)
void mbloss_main(const float* __restrict__ cout, const float* __restrict__ rout,
                 const int*   __restrict__ cls,  const float* __restrict__ regt,
                 float2* __restrict__ part, int nFullTiles, int N)
{
  __shared__ __align__(16) unsigned char smem[2 * BUF_BYTES];
  __shared__ float red[16];

  const int tid = threadIdx.x;
  float cAcc = 0.0f, rAcc = 0.0f;

  // ---- double-buffered async pipeline over full tiles ----
  long long t = blockIdx.x;
  if (t < nFullTiles) issue_tile(cout, rout, cls, regt, t * TILE, smem, tid);

  int ibuf = 0;
  for (; t < nFullTiles; t += gridDim.x) {
    long long tn = t + gridDim.x;
    if (tn < nFullTiles) {
      issue_tile(cout, rout, cls, regt, tn * TILE, smem + (ibuf ^ 1) * BUF_BYTES, tid);
      WAIT_ASYNC(4);                 // tile t's 4 ops done; tn's may remain in flight
    } else {
      WAIT_ASYNC(0);
    }
    __syncthreads();                 // publish all waves' LDS writes for tile t

    const unsigned char* buf = smem + ibuf * BUF_BYTES;
    float2 x = ((const float2*)(buf))[tid];
    float4 d = ((const float4*)(buf + 2048))[tid];
    float4 g = ((const float4*)(buf + 6144))[tid];
    int    l = ((const int*)  (buf + 10240))[tid];
    accum_row(x.x, x.y, l, d, g, cAcc, rAcc);

    __syncthreads();                 // done reading buf before it is re-filled
    ibuf ^= 1;
  }

  // ---- tail rows (N not a multiple of TILE): direct coalesced loads ----
  if (blockIdx.x == 0) {
    for (long long r = (long long)nFullTiles * TILE + tid; r < N; r += NTHREADS) {
      float2 x = ((const float2*)cout)[r];
      float4 d = ((const float4*)rout)[r];
      float4 g = ((const float4*)regt)[r];
      int    l = cls[r];
      accum_row(x.x, x.y, l, d, g, cAcc, rAcc);
    }
  }

  // ---- wave32 butterfly + cross-wave reduction ----
  for (int off = 16; off > 0; off >>= 1) {
    cAcc += __shfl_down(cAcc, off);
    rAcc += __shfl_down(rAcc, off);
  }
  const int wave = tid >> 5, lane = tid & 31;
  if (lane == 0) { red[wave] = cAcc; red[8 + wave] = rAcc; }
  __syncthreads();
  if (tid == 0) {
    float C = 0.0f, R = 0.0f;
    for (int w = 0; w < 8; ++w) { C += red[w]; R += red[8 + w]; }
    part[blockIdx.x] = make_float2(C, R);   // always written -> deterministic
  }
}

__global__ __launch_bounds__(NTHREADS)
void mbloss_final(const float2* __restrict__ part, int nb, float* __restrict__ out)
{
  __shared__ float red[16];
  const int tid = threadIdx.x;
  float c = 0.0f, r = 0.0f;
  for (int i = tid; i < nb; i += NTHREADS) { float2 p = part[i]; c += p.x; r += p.y; }
  for (int off = 16; off > 0; off >>= 1) {
    c += __shfl_down(c, off);
    r += __shfl_down(r, off);
  }
  const int wave = tid >> 5, lane = tid & 31;
  if (lane == 0) { red[wave] = c; red[8 + wave] = r; }
  __syncthreads();
  if (tid == 0) {
    float C = 0.0f, R = 0.0f;
    for (int w = 0; w < 8; ++w) { C += red[w]; R += red[8 + w]; }
    float closs = C * (1.0f / 64.0f);
    float rloss = R * (1.0f / 16.0f);
    out[0] = closs;
    out[1] = rloss;
    out[2] = closs + 10.0f * rloss;
  }
}

extern "C" void kernel_launch(void* const* d_in, const int* in_sizes, int n_in,
                              void* d_out, int out_size, void* d_ws, size_t ws_size,
                              hipStream_t stream)
{
  const float* cout = (const float*)d_in[0];   // [N,2] f32
  const float* rout = (const float*)d_in[1];   // [N,4] f32
  const int*   cls  = (const int*)  d_in[2];   // [N]   i32
  const float* regt = (const float*)d_in[3];   // [N,4] f32
  const int N = in_sizes[2];                   // cls_t has N elements

  float2* part = (float2*)d_ws;                // NBLOCKS float2 partials
  const int nFull = N / TILE;

  mbloss_main<<<NBLOCKS, NTHREADS, 0, stream>>>(cout, rout, cls, regt, part, nFull, N);
  mbloss_final<<<1, NTHREADS, 0, stream>>>(part, NBLOCKS, (float*)d_out);
}